// BaselineGNNModel_87651692577500
// MI455X (gfx1250) — compile-verified
//
#include <hip/hip_runtime.h>

// GCN inference on MI455X (gfx1250, wave32).
// N=50000 nodes, E=800000 edges, IC=H=128, OC=64, fp32 throughout.
// GEMMs: V_WMMA_F32_16X16X4_F32 with B staged in LDS (bank-conflict-free
// ds_load_b64 fragments). Edge gather/scatter is L2-resident.

#define GCN_N  50000
#define GCN_E  800000
#define GCN_H  128
#define GCN_OC 64
#define GCN_EPS 1e-5f

typedef float v2f __attribute__((ext_vector_type(2)));
typedef float v8f __attribute__((ext_vector_type(8)));

// ---------------------------------------------------------------------------
// WMMA GEMM: C[M x NT*16] = A[M x K] * B[K x NT*16] (+ bias), row-major.
// 256 threads = 8 waves/block, one 16-row tile per wave, NT col-tiles each.
// B staged in LDS, pair-interleaved:
//   lb[p*STRIDE + 2n + j] = B[2p + j][n],  p = 0..K/2-1
// Fragment for (k, half, tile t, lane lr) = 8B LDS load at
//   (k/2 + half)*STRIDE + 2*(t*16 + lr).
// STRIDE = 2*NCOL + 32 words => the two lane-halves hit disjoint bank sets.
// ---------------------------------------------------------------------------
template <int NT>
__global__ void __launch_bounds__(256)
gemm_wmma_f32(const float* __restrict__ A,
              const float* __restrict__ B,
              const float* __restrict__ bias, // NT*16 or nullptr
              float* __restrict__ C,
              int mtiles, int K)
{
    constexpr int NCOL   = NT * 16;
    constexpr int STRIDE = 2 * NCOL + 32;          // words per pair-row
    __shared__ float lb[64 * STRIDE];              // K=128 -> 64 pair-rows

    // ---- cooperative stage of B into LDS (coalesced global reads)
    {
        const int tid  = threadIdx.x;
        const int nIdx = tid % NCOL;
        const int grp  = tid / NCOL;               // 0..(256/NCOL - 1)
        const int GP   = 256 / NCOL;
        const int pairs = K >> 1;
        for (int p = grp; p < pairs; p += GP) {
            lb[p * STRIDE + 2 * nIdx]     = B[(size_t)(2 * p)     * NCOL + nIdx];
            lb[p * STRIDE + 2 * nIdx + 1] = B[(size_t)(2 * p + 1) * NCOL + nIdx];
        }
    }
    __syncthreads();

    const int wave = blockIdx.x * (blockDim.x >> 5) + (threadIdx.x >> 5);
    if (wave >= mtiles) return;                    // wave-uniform after barrier
    const int lane = threadIdx.x & 31;
    const int half = lane >> 4;                    // 0: lanes 0-15, 1: 16-31
    const int lr   = lane & 15;
    const int m0   = wave * 16;

    v8f acc[NT];
    const v8f zero = {0.f, 0.f, 0.f, 0.f, 0.f, 0.f, 0.f, 0.f};
#pragma unroll
    for (int t = 0; t < NT; ++t) acc[t] = zero;

    // A fragment base: row (m0+lr), K offset 2*half (+k). 8B aligned.
    const float* arow = A + (size_t)(m0 + lr) * K + 2 * half;

    for (int k = 0; k < K; k += 4) {
        const v2f a = *(const v2f*)(arow + k);     // {A[m][k+2h], A[m][k+2h+1]}
        const float* lbp = lb + (k / 2 + half) * STRIDE + 2 * lr;
        v2f bf[NT];
#pragma unroll
        for (int t = 0; t < NT; ++t)               // batch the ds loads
            bf[t] = *(const v2f*)(lbp + t * 32);
#pragma unroll
        for (int t = 0; t < NT; ++t)
            acc[t] = __builtin_amdgcn_wmma_f32_16x16x4_f32(
                false, a, false, bf[t], (short)0, acc[t], false, false);
    }

#pragma unroll
    for (int t = 0; t < NT; ++t) {
#pragma unroll
        for (int r = 0; r < 8; ++r) {
            const int m = m0 + r + 8 * half;       // C VGPR-r row mapping
            const int n = t * 16 + lr;
            float v = acc[t][r];
            if (bias) v += bias[n];
            C[(size_t)m * NCOL + n] = v;
        }
    }
}

// ---------------------------------------------------------------------------
// Degree / normalization
// ---------------------------------------------------------------------------
__global__ void deg_init(float* __restrict__ deg, int n)
{
    int i = blockIdx.x * blockDim.x + threadIdx.x;
    if (i < n) deg[i] = 1.0f;                      // self-loop (+1)
}

__global__ void deg_count(const long long* __restrict__ dst,
                          float* __restrict__ deg, int e)
{
    int i = blockIdx.x * blockDim.x + threadIdx.x;
    if (i < e) atomicAdd(&deg[(int)dst[i]], 1.0f);
}

__global__ void deg_rsqrt(float* __restrict__ deg, int n)
{
    int i = blockIdx.x * blockDim.x + threadIdx.x;
    if (i < n) deg[i] = rsqrtf(deg[i]);            // dinv = deg^-1/2
}

// ---------------------------------------------------------------------------
// agg[i,:] = hw[i,:] * dinv[i]^2  (self-loop term; also initializes agg)
// ---------------------------------------------------------------------------
__global__ void self_init(const float* __restrict__ hw,
                          const float* __restrict__ dinv,
                          float* __restrict__ agg, int nvec)
{
    int t = blockIdx.x * blockDim.x + threadIdx.x;
    if (t >= nvec) return;
    const int row = t >> 5, c4 = t & 31;
    const float di = dinv[row];
    const float s  = di * di;
    float4 v = ((const float4*)hw)[(size_t)row * 32 + c4];
    v.x *= s; v.y *= s; v.z *= s; v.w *= s;
    ((float4*)agg)[(size_t)row * 32 + c4] = v;
}

// ---------------------------------------------------------------------------
// Edge scatter: one wave per edge, 32 lanes x float4 = 128 channels.
// agg[dst] += hw[src] * dinv[src]*dinv[dst]  (L2-resident atomics)
// ---------------------------------------------------------------------------
__global__ void scatter_edges(const long long* __restrict__ src,
                              const long long* __restrict__ dst,
                              const float* __restrict__ dinv,
                              const float* __restrict__ hw,
                              float* __restrict__ agg, int e)
{
    int t = blockIdx.x * blockDim.x + threadIdx.x;
    const int edge = t >> 5;
    if (edge >= e) return;
    const int lane = t & 31;
    const int s = (int)src[edge];
    const int d = (int)dst[edge];
    const float coef = dinv[s] * dinv[d];
    float4 v = ((const float4*)(hw + (size_t)s * GCN_H))[lane];
    float* ad = agg + (size_t)d * GCN_H + lane * 4;
    atomicAdd(ad + 0, v.x * coef);
    atomicAdd(ad + 1, v.y * coef);
    atomicAdd(ad + 2, v.z * coef);
    atomicAdd(ad + 3, v.w * coef);
}

// ---------------------------------------------------------------------------
// Fused conv-bias + BatchNorm(eval) + ReLU, in place over agg.
// ---------------------------------------------------------------------------
__global__ void bias_bn_relu(float* __restrict__ agg,
                             const float* __restrict__ bw,
                             const float* __restrict__ g,
                             const float* __restrict__ be,
                             const float* __restrict__ m,
                             const float* __restrict__ v,
                             int nvec)
{
    int t = blockIdx.x * blockDim.x + threadIdx.x;
    if (t >= nvec) return;
    const int row = t >> 5, c4 = t & 31, c = c4 * 4;
    float4 a = ((const float4*)agg)[(size_t)row * 32 + c4];
    float o[4] = {a.x, a.y, a.z, a.w};
#pragma unroll
    for (int j = 0; j < 4; ++j) {
        float val = (o[j] + bw[c + j] - m[c + j]) * rsqrtf(v[c + j] + GCN_EPS)
                        * g[c + j] + be[c + j];
        o[j] = val > 0.0f ? val : 0.0f;
    }
    a.x = o[0]; a.y = o[1]; a.z = o[2]; a.w = o[3];
    ((float4*)agg)[(size_t)row * 32 + c4] = a;
}

// ---------------------------------------------------------------------------
extern "C" void kernel_launch(void* const* d_in, const int* in_sizes, int n_in,
                              void* d_out, int out_size, void* d_ws, size_t ws_size,
                              hipStream_t stream)
{
    const float*     x   = (const float*)d_in[0];
    const long long* ei  = (const long long*)d_in[1];   // int64 [2,E]
    const float* W1  = (const float*)d_in[2];
    const float* bw1 = (const float*)d_in[3];
    const float* W2  = (const float*)d_in[4];
    const float* bw2 = (const float*)d_in[5];
    const float* W3  = (const float*)d_in[6];
    const float* bw3 = (const float*)d_in[7];
    const float* g1  = (const float*)d_in[8];
    const float* be1 = (const float*)d_in[9];
    const float* m1  = (const float*)d_in[10];
    const float* v1  = (const float*)d_in[11];
    const float* g2  = (const float*)d_in[12];
    const float* be2 = (const float*)d_in[13];
    const float* m2  = (const float*)d_in[14];
    const float* v2  = (const float*)d_in[15];
    const float* g3  = (const float*)d_in[16];
    const float* be3 = (const float*)d_in[17];
    const float* m3  = (const float*)d_in[18];
    const float* v3  = (const float*)d_in[19];
    const float* Wh  = (const float*)d_in[20];
    const float* bh  = (const float*)d_in[21];
    float* out = (float*)d_out;

    const int N = GCN_N, E = GCN_E, H = GCN_H;

    // workspace: dinv[N] | hw[N*H] | agg[N*H]  (~51.5 MB)
    float* dinv = (float*)d_ws;
    float* hw   = dinv + N;
    float* agg  = hw + (size_t)N * H;

    const long long* src = ei;
    const long long* dst = ei + E;

    // ---- degrees + symmetric normalization
    deg_init <<<(N + 255) / 256, 256, 0, stream>>>(dinv, N);
    deg_count<<<(E + 255) / 256, 256, 0, stream>>>(dst, dinv, E);
    deg_rsqrt<<<(N + 255) / 256, 256, 0, stream>>>(dinv, N);

    const int mtiles = N / 16;                       // 3125 (N % 16 == 0)
    const dim3 gBlk(256);                            // 8 waves / block
    const dim3 gGrid((mtiles + 7) / 8);
    const int nvec = N * 32;                         // float4s per feature map
    const dim3 eBlk(256);
    const dim3 eGrid((unsigned)(((long long)E * 32 + 255) / 256));
    const dim3 nGrid((nvec + 255) / 256);

    // ---- layer 1: input x
    gemm_wmma_f32<8><<<gGrid, gBlk, 0, stream>>>(x, W1, nullptr, hw, mtiles, H);
    self_init    <<<nGrid, eBlk, 0, stream>>>(hw, dinv, agg, nvec);
    scatter_edges<<<eGrid, eBlk, 0, stream>>>(src, dst, dinv, hw, agg, E);
    bias_bn_relu <<<nGrid, eBlk, 0, stream>>>(agg, bw1, g1, be1, m1, v1, nvec);

    // ---- layer 2 (agg holds h1)
    gemm_wmma_f32<8><<<gGrid, gBlk, 0, stream>>>(agg, W2, nullptr, hw, mtiles, H);
    self_init    <<<nGrid, eBlk, 0, stream>>>(hw, dinv, agg, nvec);
    scatter_edges<<<eGrid, eBlk, 0, stream>>>(src, dst, dinv, hw, agg, E);
    bias_bn_relu <<<nGrid, eBlk, 0, stream>>>(agg, bw2, g2, be2, m2, v2, nvec);

    // ---- layer 3
    gemm_wmma_f32<8><<<gGrid, gBlk, 0, stream>>>(agg, W3, nullptr, hw, mtiles, H);
    self_init    <<<nGrid, eBlk, 0, stream>>>(hw, dinv, agg, nvec);
    scatter_edges<<<eGrid, eBlk, 0, stream>>>(src, dst, dinv, hw, agg, E);
    bias_bn_relu <<<nGrid, eBlk, 0, stream>>>(agg, bw3, g3, be3, m3, v3, nvec);

    // ---- head: out = h3 @ Wh + bh  (N x 64)
    gemm_wmma_f32<4><<<gGrid, gBlk, 0, stream>>>(agg, Wh, bh, out, mtiles, H);
}